// GNNModel_1958505087710
// MI455X (gfx1250) — compile-verified
//
#include <hip/hip_runtime.h>
#include <hip/hip_bf16.h>
#include <math.h>

typedef __attribute__((ext_vector_type(2))) float v2f;
typedef __attribute__((ext_vector_type(8))) float v8f;

// ---------------------------------------------------------------------------
// Small elementwise kernels
// ---------------------------------------------------------------------------
__global__ void k_fill(float* __restrict__ p, float v, int n) {
    int i = blockIdx.x * blockDim.x + threadIdx.x;
    if (i < n) p[i] = v;
}

__global__ void k_deg_edges(const int* __restrict__ dst, float* __restrict__ deg, int E) {
    int e = blockIdx.x * blockDim.x + threadIdx.x;
    if (e < E) atomicAdd(&deg[dst[e]], 1.0f);
}

__global__ void k_rsqrt(float* __restrict__ p, int n) {
    int i = blockIdx.x * blockDim.x + threadIdx.x;
    if (i < n) p[i] = rsqrtf(p[i]);
}

// out[i][f] = in[i][f] * dinv[i]^2   (self-loop message; also zero-initializes out)
__global__ void k_selfloop(const float* __restrict__ in, const float* __restrict__ dinv,
                           float* __restrict__ out, long long total, int logF) {
    long long idx = (long long)blockIdx.x * blockDim.x + threadIdx.x;
    if (idx >= total) return;
    int i = (int)(idx >> logF);
    float dv = dinv[i];
    out[idx] = in[idx] * dv * dv;
}

// ---------------------------------------------------------------------------
// Edge propagation: one wave32 per edge. Gather source row (L2-resident),
// scale by dinv[s]*dinv[d], scatter with coalesced f32 atomics.
// ---------------------------------------------------------------------------
__global__ void k_scatter(const float* __restrict__ in, float* __restrict__ out,
                          const int* __restrict__ src, const int* __restrict__ dst,
                          const float* __restrict__ dinv, int E, int F) {
    int wid  = (blockIdx.x * blockDim.x + threadIdx.x) >> 5;   // wave-uniform
    int lane = threadIdx.x & 31;
    if (wid >= E) return;
    int s = src[wid];
    int d = dst[wid];
    float w = dinv[s] * dinv[d];
    const float* pin  = in  + (size_t)s * F;
    float*       pout = out + (size_t)d * F;
    for (int c = lane * 4; c < F; c += 128) {
        float4 v = *(const float4*)(pin + c);
        atomicAdd(pout + c + 0, v.x * w);
        atomicAdd(pout + c + 1, v.y * w);
        atomicAdd(pout + c + 2, v.z * w);
        atomicAdd(pout + c + 3, v.w * w);
    }
}

// ---------------------------------------------------------------------------
// fp32 WMMA GEMM: D[N x 256] = relu(A[N x K] @ B[K x 256] + bias)
// One wave computes one 16x16 tile with V_WMMA_F32_16X16X4_F32.
// A layout : lanes 0-15 -> M=lane, K={k,k+1}; lanes 16-31 -> M=lane-16, K={k+2,k+3}
// B layout : mirrored (lane%16 = N column, lane/16 selects K pair)
// C/D      : VGPR i -> M = i + 8*(lane/16), N = lane%16
// ---------------------------------------------------------------------------
template <int K>
__global__ __launch_bounds__(256) void k_gemm_bias_relu(const float* __restrict__ A,
                                                        const float* __restrict__ B,
                                                        const float* __restrict__ bias,
                                                        float* __restrict__ D,
                                                        int nTiles) {
    int wid  = (blockIdx.x * blockDim.x + threadIdx.x) >> 5;   // wave-uniform
    int lane = threadIdx.x & 31;
    if (wid >= nTiles) return;                                  // wave-uniform branch
    int trow = wid >> 4;        // 16 column tiles (256 cols)
    int tcol = wid & 15;
    int lo = lane & 15;
    int hi = lane >> 4;

    const float* Ar = A + ((size_t)trow * 16 + lo) * K;         // row M = lo
    const float* Bc = B + tcol * 16 + lo;                       // col N = lo

    v8f c = {};
#pragma unroll 4
    for (int k = 0; k < K; k += 4) {
        v2f a, b;
        a.x = Ar[k + 2 * hi];
        a.y = Ar[k + 2 * hi + 1];
        b.x = Bc[(size_t)(k + 2 * hi) * 256];
        b.y = Bc[(size_t)(k + 2 * hi + 1) * 256];
        c = __builtin_amdgcn_wmma_f32_16x16x4_f32(false, a, false, b,
                                                  (short)0, c, false, false);
    }

    float bval = bias[tcol * 16 + lo];
    float* Dp = D + ((size_t)trow * 16 + (size_t)hi * 8) * 256 + tcol * 16 + lo;
#pragma unroll
    for (int i = 0; i < 8; i++) {
        float v = c[i] + bval;
        Dp[(size_t)i * 256] = fmaxf(v, 0.0f);
    }
}

// ---------------------------------------------------------------------------
// Mean-pool accumulation: one wave per node scatters its 256-dim row into
// pooled[batch[i]] ; lane 0 counts the node.
// ---------------------------------------------------------------------------
__global__ void k_pool(const float* __restrict__ h, const int* __restrict__ batch,
                       float* __restrict__ pooled, float* __restrict__ counts, int N) {
    int wid  = (blockIdx.x * blockDim.x + threadIdx.x) >> 5;
    int lane = threadIdx.x & 31;
    if (wid >= N) return;
    int g = batch[wid];
    const float* p = h + (size_t)wid * 256;
    float* q = pooled + (size_t)g * 256;
    for (int c = lane * 4; c < 256; c += 128) {
        float4 v = *(const float4*)(p + c);
        atomicAdd(q + c + 0, v.x);
        atomicAdd(q + c + 1, v.y);
        atomicAdd(q + c + 2, v.z);
        atomicAdd(q + c + 3, v.w);
    }
    if (lane == 0) atomicAdd(&counts[g], 1.0f);
}

// ---------------------------------------------------------------------------
// Head: logits = (pooled/cnt) @ fc_w + fc_b, then log_softmax. One thread/graph.
// ---------------------------------------------------------------------------
__global__ void k_head(const float* __restrict__ pooled, const float* __restrict__ counts,
                       const float* __restrict__ fcw, const float* __restrict__ fcb,
                       float* __restrict__ out, int G, int H, int C) {
    int g = blockIdx.x * blockDim.x + threadIdx.x;
    if (g >= G) return;
    float inv = 1.0f / fmaxf(counts[g], 1.0f);
    float logits[8];
    for (int c = 0; c < C; c++) {
        float acc = fcb[c];
        const float* pg = pooled + (size_t)g * H;
        for (int f = 0; f < H; f++) acc += pg[f] * inv * fcw[(size_t)f * C + c];
        logits[c] = acc;
    }
    float m = logits[0];
    for (int c = 1; c < C; c++) m = fmaxf(m, logits[c]);
    float sum = 0.0f;
    for (int c = 0; c < C; c++) sum += expf(logits[c] - m);
    float lse = m + logf(sum);
    for (int c = 0; c < C; c++) out[(size_t)g * C + c] = logits[c] - lse;
}

// ---------------------------------------------------------------------------
extern "C" void kernel_launch(void* const* d_in, const int* in_sizes, int n_in,
                              void* d_out, int out_size, void* d_ws, size_t ws_size,
                              hipStream_t stream) {
    const float* x    = (const float*)d_in[0];   // [N, 128]
    const int*   ei   = (const int*)d_in[1];     // [2, E]
    const int*   batch= (const int*)d_in[2];     // [N]
    const float* W1   = (const float*)d_in[3];   // [128, 256]
    const float* b1   = (const float*)d_in[4];   // [256]
    const float* W2   = (const float*)d_in[5];   // [256, 256]
    const float* b2   = (const float*)d_in[6];   // [256]
    const float* fcw  = (const float*)d_in[7];   // [256, 2]
    const float* fcb  = (const float*)d_in[8];   // [2]

    const int N = in_sizes[2];                   // 100000
    const int E = in_sizes[1] / 2;               // 3200000
    const int F = in_sizes[0] / N;               // 128
    const int H = in_sizes[4];                   // 256
    const int C = in_sizes[8];                   // 2
    const int G = 64;

    const int* src = ei;
    const int* dst = ei + E;

    // workspace layout
    float* buf0   = (float*)d_ws;                // N*256
    float* buf1   = buf0 + (size_t)N * H;        // N*256
    float* dinv   = buf1 + (size_t)N * H;        // N   (deg, then rsqrt in place)
    float* pooled = dinv + N;                    // G*H
    float* counts = pooled + (size_t)G * H;      // G

    const int T = 256;
    auto blocks = [](long long work, int t) { return (int)((work + t - 1) / t); };

    // 1) degrees (self-loop => start at 1), then dinv = rsqrt(deg)
    k_fill<<<blocks(N, T), T, 0, stream>>>(dinv, 1.0f, N);
    k_deg_edges<<<blocks(E, T), T, 0, stream>>>(dst, dinv, E);
    k_rsqrt<<<blocks(N, T), T, 0, stream>>>(dinv, N);

    // 2) layer 1: propagate x first (128 features), then GEMM  (P x) W1
    k_selfloop<<<blocks((long long)N * F, T), T, 0, stream>>>(x, dinv, buf0, (long long)N * F, 7);
    k_scatter<<<blocks((long long)E * 32, T), T, 0, stream>>>(x, buf0, src, dst, dinv, E, F);
    {
        int nTiles = (N / 16) * 16;              // 6250 row tiles * 16 col tiles
        k_gemm_bias_relu<128><<<blocks((long long)nTiles * 32, T), T, 0, stream>>>(
            buf0, W1, b1, buf1, nTiles);
    }

    // 3) layer 2: propagate h1 (256 features), then GEMM
    k_selfloop<<<blocks((long long)N * H, T), T, 0, stream>>>(buf1, dinv, buf0, (long long)N * H, 8);
    k_scatter<<<blocks((long long)E * 32, T), T, 0, stream>>>(buf1, buf0, src, dst, dinv, E, H);
    {
        int nTiles = (N / 16) * 16;
        k_gemm_bias_relu<256><<<blocks((long long)nTiles * 32, T), T, 0, stream>>>(
            buf0, W2, b2, buf1, nTiles);
    }

    // 4) mean pool + head
    k_fill<<<blocks(G * H + G, T), T, 0, stream>>>(pooled, 0.0f, G * H + G);  // pooled+counts contiguous
    k_pool<<<blocks((long long)N * 32, T), T, 0, stream>>>(buf1, batch, pooled, counts, N);
    k_head<<<1, 64, 0, stream>>>(pooled, counts, fcw, fcb, (float*)d_out, G, H, C);
}